// Channel_network_32710470926817
// MI455X (gfx1250) — compile-verified
//
#include <hip/hip_runtime.h>
#include <hip/hip_bf16.h>

typedef float v2f __attribute__((ext_vector_type(2)));
typedef float v4f __attribute__((ext_vector_type(4)));
typedef float v8f __attribute__((ext_vector_type(8)));

#define C_DIM 256
#define B_DIM 8
#define HW    (256 * 256)

// ---------------------------------------------------------------------------
// Kernel 1: gate = 1 + softmax(LeakyReLU(semantic @ W1^T + b1) @ W2^T + b2)
// One workgroup, 512 threads = 16 wave32s. Each wave computes one 16-wide
// N-tile of the 256 output channels via V_WMMA_F32_16X16X4_F32 (fp32-exact).
// Batch M=8 is zero-padded to 16.
// ---------------------------------------------------------------------------
__global__ __launch_bounds__(512) void gate_kernel(
    const float* __restrict__ semantic,  // [8,256]
    const float* __restrict__ W1,        // [256,256]
    const float* __restrict__ b1,        // [256]
    const float* __restrict__ W2,        // [256,256]
    const float* __restrict__ b2,        // [256]
    float* __restrict__ gate)            // [8,256] out: 1 + softmax(...)
{
    __shared__ float Alds[16 * C_DIM];   // A operand (semantic, then logits)
    __shared__ float Hlds[16 * C_DIM];   // hidden activations

    const int tid = threadIdx.x;

    // Stage semantic into LDS, zero-pad rows 8..15.
    for (int i = tid; i < 16 * C_DIM; i += 512) {
        const int m = i >> 8;
        const int k = i & (C_DIM - 1);
        Alds[i] = (m < B_DIM) ? semantic[m * C_DIM + k] : 0.0f;
    }
    __syncthreads();

    const int wave  = tid >> 5;          // 0..15 -> N tile
    const int lane  = tid & 31;
    const int n0    = wave * 16;
    const int mRow  = lane & 15;         // A-fragment row (M) / B col within tile
    const int kOff  = (lane >> 4) * 2;   // K sub-offset per half-wave
    const int nCol  = n0 + mRow;         // global N handled by this lane for B/C/D
    const int mBase = (lane >> 4) * 8;   // C/D fragment M base

    // ---- GEMM1: H = semantic @ W1^T ----
    v8f acc = {};
    for (int k = 0; k < C_DIM; k += 4) {
        v2f a, b;
        a.x = Alds[mRow * C_DIM + k + kOff];
        a.y = Alds[mRow * C_DIM + k + kOff + 1];
        b.x = W1[nCol * C_DIM + k + kOff];       // B[k][n] = W1[n][k]
        b.y = W1[nCol * C_DIM + k + kOff + 1];
        acc = __builtin_amdgcn_wmma_f32_16x16x4_f32(
            /*neg_a=*/false, a, /*neg_b=*/false, b,
            /*c_mod=*/(short)0, acc, /*reuse_a=*/false, /*reuse_b=*/false);
    }

    // Bias + LeakyReLU(0.1), scatter C/D fragment into Hlds.
    {
        const float bias = b1[nCol];
#pragma unroll
        for (int v = 0; v < 8; ++v) {
            float h = acc[v] + bias;
            h = (h > 0.0f) ? h : 0.1f * h;
            Hlds[(mBase + v) * C_DIM + nCol] = h;
        }
    }
    __syncthreads();

    // ---- GEMM2: logits = H @ W2^T ----
    v8f acc2 = {};
    for (int k = 0; k < C_DIM; k += 4) {
        v2f a, b;
        a.x = Hlds[mRow * C_DIM + k + kOff];
        a.y = Hlds[mRow * C_DIM + k + kOff + 1];
        b.x = W2[nCol * C_DIM + k + kOff];
        b.y = W2[nCol * C_DIM + k + kOff + 1];
        acc2 = __builtin_amdgcn_wmma_f32_16x16x4_f32(
            false, a, false, b, (short)0, acc2, false, false);
    }

    // Bias, stash logits into Alds (semantic no longer needed).
    {
        const float bias = b2[nCol];
#pragma unroll
        for (int v = 0; v < 8; ++v) {
            Alds[(mBase + v) * C_DIM + nCol] = acc2[v] + bias;
        }
    }
    __syncthreads();

    // ---- Row softmax over C for the 8 real batch rows; write 1 + w ----
    if (tid < B_DIM) {
        const float* row = &Alds[tid * C_DIM];
        float mx = -__builtin_inff();
        for (int n = 0; n < C_DIM; ++n) mx = fmaxf(mx, row[n]);
        float s = 0.0f;
        for (int n = 0; n < C_DIM; ++n) s += __expf(row[n] - mx);
        const float inv = 1.0f / s;
        for (int n = 0; n < C_DIM; ++n)
            gate[tid * C_DIM + n] = 1.0f + __expf(row[n] - mx) * inv;
    }
}

// ---------------------------------------------------------------------------
// Kernel 2: out[b,c,:,:] = x[b,c,:,:] * gate[b,c]
// Bandwidth-bound streaming pass: non-temporal B128 loads/stores,
// gate fetched once per block via a uniform (scalar) load.
// grid = (16, B*C), block = 256; each thread handles 4 float4s.
// ---------------------------------------------------------------------------
__global__ __launch_bounds__(256) void scale_kernel(
    const v4f* __restrict__ x4,
    const float* __restrict__ gate,
    v4f* __restrict__ out4)
{
    const int plane4 = HW / 4;                       // 16384 float4 per plane
    const unsigned bc = blockIdx.y;                  // (b*C + c), uniform
    const float g = gate[bc];                        // scalar (s_load) path
    const long long base = (long long)bc * plane4;

    for (int j = blockIdx.x * blockDim.x + threadIdx.x; j < plane4;
         j += gridDim.x * blockDim.x) {
        v4f v = __builtin_nontemporal_load(&x4[base + j]);
        v *= g;
        __builtin_nontemporal_store(v, &out4[base + j]);
    }
}

extern "C" void kernel_launch(void* const* d_in, const int* in_sizes, int n_in,
                              void* d_out, int out_size, void* d_ws, size_t ws_size,
                              hipStream_t stream) {
    const float* x        = (const float*)d_in[0];
    const float* semantic = (const float*)d_in[1];
    const float* W1       = (const float*)d_in[2];
    const float* b1       = (const float*)d_in[3];
    const float* W2       = (const float*)d_in[4];
    const float* b2       = (const float*)d_in[5];
    float* out            = (float*)d_out;
    float* gate           = (float*)d_ws;            // [B*C] = 2048 floats

    gate_kernel<<<1, 512, 0, stream>>>(semantic, W1, b1, W2, b2, gate);

    dim3 grid(16, B_DIM * C_DIM);
    scale_kernel<<<grid, 256, 0, stream>>>((const v4f*)x, gate, (v4f*)out);
}